// DotProductPredictor_33122787786913
// MI455X (gfx1250) — compile-verified
//
#include <hip/hip_runtime.h>
#include <hip/hip_bf16.h>

// CDNA5 f32 WMMA fragment types
typedef __attribute__((ext_vector_type(2))) float v2f;   // A/B frag for 16x16x4 f32
typedef __attribute__((ext_vector_type(8))) float v8f;   // 16x16 f32 C/D frag

// One wave32 computes 16 edge dot-products (d=128) via 32 chained
// V_WMMA_F32_16X16X4_F32 ops; results are the diagonal of the 16x16 accumulator.
//   A[m][k]  = h[src[e0+m]][4*ch + k]   (m = lane&15, k split across lane halves)
//   B[k][n]  = h[dst[e0+n]][4*ch + k]
//   D[e][e] += dot chunk  ->  full 128-d dot after 32 chunks.
__global__ __launch_bounds__(256) void edge_dot_wmma_f32(
    const float* __restrict__ h,
    const int*   __restrict__ src,
    const int*   __restrict__ dst,
    float*       __restrict__ out,
    int n_edges)
{
    const int lane  = threadIdx.x & 31;
    const int wave  = blockIdx.x * (blockDim.x >> 5) + (threadIdx.x >> 5);
    const int e0    = wave * 16;
    if (e0 >= n_edges) return;          // wave-uniform: EXEC stays all-1s for WMMA

    const int m    = lane & 15;         // edge-in-wave (row for A, col for B)
    const int half = lane >> 4;         // 0 -> k_local {0,1}, 1 -> k_local {2,3}

    // Gather row base pointers (random gather; h is L2-resident: 51.2MB < 192MB L2)
    const long long sIdx = (long long)src[e0 + m];
    const long long dIdx = (long long)dst[e0 + m];
    const float* __restrict__ srow = h + sIdx * 128;
    const float* __restrict__ drow = h + dIdx * 128;

    v8f c = {};
#pragma unroll
    for (int ch = 0; ch < 32; ++ch) {
        const int k = ch * 4 + half * 2;                 // this lane's K pair
        v2f a = *(const v2f*)(srow + k);                 // global_load_b64
        v2f b = *(const v2f*)(drow + k);                 // global_load_b64
        // 8 args: (neg_a, A, neg_b, B, c_mod, C, reuse_a, reuse_b)
        c = __builtin_amdgcn_wmma_f32_16x16x4_f32(
                false, a, false, b, (short)0, c, false, false);
    }

    // Diagonal extraction per documented 16x16 f32 C/D layout:
    //   e in [0,8):  lane e,     VGPR e
    //   e in [8,16): lane e+16,  VGPR e-8
    // In both cases the VGPR index is (lane & 7).
    const int sel = lane & 7;
    float val = c[0];
    if (sel == 1) val = c[1];
    if (sel == 2) val = c[2];
    if (sel == 3) val = c[3];
    if (sel == 4) val = c[4];
    if (sel == 5) val = c[5];
    if (sel == 6) val = c[6];
    if (sel == 7) val = c[7];

    if (lane < 8)        out[e0 + lane]        = val;   // edges 0..7
    else if (lane >= 24) out[e0 + (lane - 16)] = val;   // edges 8..15
}

extern "C" void kernel_launch(void* const* d_in, const int* in_sizes, int n_in,
                              void* d_out, int out_size, void* d_ws, size_t ws_size,
                              hipStream_t stream) {
    const float* h   = (const float*)d_in[0];
    const int*   src = (const int*)  d_in[1];
    const int*   dst = (const int*)  d_in[2];
    float*       out = (float*)      d_out;

    const int n_edges = in_sizes[1];              // 640000
    const int edges_per_block = (256 / 32) * 16;  // 8 waves * 16 edges = 128
    const int grid = (n_edges + edges_per_block - 1) / edges_per_block;

    edge_dot_wmma_f32<<<grid, 256, 0, stream>>>(h, src, dst, out, n_edges);
}